// SemiConv2d_1606317768738
// MI455X (gfx1250) — compile-verified
//
#include <hip/hip_runtime.h>

// Problem constants (shapes fixed by the reference).
#define B    16
#define C    32
#define OC   32
#define H    256
#define W    256
#define KS   5
#define PAD  2
#define HWSZ (H * W)          // 65536
#define PW   (W + 2 * PAD)    // 260 padded width/height
#define PIMG (PW * PW)        // 67600 floats per padded image
#define TILE 32               // output tile (32x32)
#define LTILE (TILE + KS - 1) // 36 input tile

typedef unsigned int u32x4 __attribute__((ext_vector_type(4)));
typedef int          i32x8 __attribute__((ext_vector_type(8)));
typedef int          i32x4 __attribute__((ext_vector_type(4)));
typedef float        f32x4 __attribute__((ext_vector_type(4)));

#ifndef USE_TDM
#define USE_TDM 1
#endif
#if USE_TDM && defined(__gfx1250__) && \
    __has_builtin(__builtin_amdgcn_tensor_load_to_lds) && \
    __has_builtin(__builtin_amdgcn_s_wait_tensorcnt)
#define HAVE_TDM 1
#else
#define HAVE_TDM 0
#endif

#define NEG_INF (-__builtin_inff())

// ---------------- Pass 2: max-min conv over a 32x32 tile ----------------
// (Placed first in the file so the returned asm snippet shows this kernel.)
__global__ __launch_bounds__(256) void k_conv(const float* __restrict__ xpad,
                                              const float* __restrict__ kern,
                                              float* __restrict__ out) {
    __shared__ __align__(16) float tile[LTILE * LTILE];

    const int bx = blockIdx.x, by = blockIdx.y, b = blockIdx.z;
    const int tX = bx * TILE, tY = by * TILE;
    const float* src = xpad + (size_t)b * PIMG + (size_t)tY * PW + tX;

#if HAVE_TDM
    // Tensor Data Mover: DMA the 36x36 fp32 tile (row stride 260) into LDS.
    // Tile is always fully in-bounds inside the padded buffer => no OOB.
    if (threadIdx.x < 32) {           // one wave issues the DMA
        unsigned long long ga = (unsigned long long)(uintptr_t)src;
        u32x4 g0;
        g0[0] = 1u;                                      // count = 1 valid D#
        g0[1] = (unsigned)(uintptr_t)tile;               // lds_addr (byte offset)
        g0[2] = (unsigned)ga;                            // global_addr[31:0]
        g0[3] = ((unsigned)(ga >> 32) & 0x01FFFFFFu)     // global_addr[56:32]
                | 0x80000000u;                           // type = 2 ("image")
        i32x8 g1;
        g1[0] = 0x00020000;                      // data_size = 2 (4 bytes)
        g1[1] = (int)(260u << 16);               // tensor_dim0 = 260
        g1[2] = (int)((unsigned)(B * PW) << 16); // tensor_dim1 = 4160 (whole buffer)
        g1[3] = (int)((unsigned)LTILE << 16);    // tile_dim0 = 36
        g1[4] = LTILE;                           // tile_dim1 = 36, tile_dim2 = 0
        g1[5] = PW;                              // tensor_dim0_stride = 260
        g1[6] = 0;
        g1[7] = 0;
        i32x4 gz; gz[0] = gz[1] = gz[2] = gz[3] = 0;     // groups 2/3 unused (2D)
#if __clang_major__ >= 23
        i32x8 gz8;
        gz8[0]=gz8[1]=gz8[2]=gz8[3]=gz8[4]=gz8[5]=gz8[6]=gz8[7]=0;
        __builtin_amdgcn_tensor_load_to_lds(g0, g1, gz, gz, gz8, 0);
#else
        __builtin_amdgcn_tensor_load_to_lds(g0, g1, gz, gz, 0);
#endif
        __builtin_amdgcn_s_wait_tensorcnt(0);
    }
#else
    // Fallback: cooperative fill (padded buffer => no bounds checks).
    for (int idx = threadIdx.x; idx < LTILE * LTILE; idx += 256) {
        int r = idx / LTILE;
        int c = idx - r * LTILE;
        tile[idx] = src[(size_t)r * PW + c];
    }
#endif
    __syncthreads();

    // Thread -> 4 adjacent output pixels (row ty, cols x0..x0+3), all 32 oc.
    const int t  = threadIdx.x;
    const int ty = t >> 3;          // 0..31
    const int x0 = (t & 7) << 2;    // 0,4,...,28

    // 5x8 register window covering the 5x5 patches of 4 adjacent pixels.
    float row[KS][8];
#pragma unroll
    for (int dy = 0; dy < KS; ++dy) {
        const f32x4 a  = *(const f32x4*)&tile[(ty + dy) * LTILE + x0];
        const f32x4 c4 = *(const f32x4*)&tile[(ty + dy) * LTILE + x0 + 4];
        row[dy][0] = a[0];  row[dy][1] = a[1];  row[dy][2] = a[2];  row[dy][3] = a[3];
        row[dy][4] = c4[0]; row[dy][5] = c4[1]; row[dy][6] = c4[2]; row[dy][7] = c4[3];
    }

    const int gy = tY + ty;
    const int gx = tX + x0;
    float* op = out + (size_t)b * OC * HWSZ + (size_t)gy * W + gx;

    // oc loop unrolled by 2 -> 8 independent accumulator chains per thread
    // (better VOPD dual-issue pairing), 50 scalar weight loads per iter.
#pragma unroll 2
    for (int oc = 0; oc < OC; ++oc) {
        const float* kk = kern + oc * (KS * KS);  // wave-uniform -> s_loads
        float a0 = NEG_INF, a1 = NEG_INF, a2 = NEG_INF, a3 = NEG_INF;
#pragma unroll
        for (int dy = 0; dy < KS; ++dy) {
#pragma unroll
            for (int dx = 0; dx < KS; ++dx) {
                const float kv = kk[dy * KS + dx];
                a0 = fmaxf(a0, fminf(row[dy][dx    ], kv));
                a1 = fmaxf(a1, fminf(row[dy][dx + 1], kv));
                a2 = fmaxf(a2, fminf(row[dy][dx + 2], kv));
                a3 = fmaxf(a3, fminf(row[dy][dx + 3], kv));
            }
        }
        f32x4 r; r[0] = a0; r[1] = a1; r[2] = a2; r[3] = a3;
        // Non-temporal: output streams through L2 once; keep xpad resident.
        __builtin_nontemporal_store(r, (f32x4*)(op + (size_t)oc * HWSZ));
    }
}

// ---------------- Pass 0: fill padded xmax buffer with -inf ----------------
__global__ void k_fill(float* __restrict__ p) {
    int i = blockIdx.x * blockDim.x + threadIdx.x;
    if (i < (B * PIMG) / 4) {
        f32x4 v;
        v[0] = NEG_INF; v[1] = NEG_INF; v[2] = NEG_INF; v[3] = NEG_INF;
        ((f32x4*)p)[i] = v;
    }
}

// ---------------- Pass 1: channel max -> padded interior ----------------
// One thread = 4 consecutive pixels. Coalesced float4 streaming reads over
// the 32 channel planes (134 MB total, non-temporal), 4.3 MB regular write
// so xpad stays L2-resident for pass 2.
__global__ __launch_bounds__(256) void k_cmax(const float* __restrict__ in,
                                              float* __restrict__ xpad) {
    int i   = blockIdx.x * blockDim.x + threadIdx.x; // 0 .. B*H*W/4-1
    int pix = i << 2;
    int b   = pix >> 16;          // / (H*W)
    int rem = pix & (HWSZ - 1);
    const f32x4* p = (const f32x4*)(in + (size_t)b * C * HWSZ + rem);
    f32x4 m = __builtin_nontemporal_load(p);
#pragma unroll
    for (int c = 1; c < C; ++c) {
        f32x4 v = __builtin_nontemporal_load(p + (size_t)c * (HWSZ / 4));
        m[0] = fmaxf(m[0], v[0]);
        m[1] = fmaxf(m[1], v[1]);
        m[2] = fmaxf(m[2], v[2]);
        m[3] = fmaxf(m[3], v[3]);
    }
    int y = rem >> 8;
    int x = rem & (W - 1);
    float* d = xpad + (size_t)b * PIMG + (size_t)(y + PAD) * PW + (x + PAD);
    d[0] = m[0]; d[1] = m[1]; d[2] = m[2]; d[3] = m[3];
}

extern "C" void kernel_launch(void* const* d_in, const int* in_sizes, int n_in,
                              void* d_out, int out_size, void* d_ws, size_t ws_size,
                              hipStream_t stream) {
    (void)in_sizes; (void)n_in; (void)out_size; (void)ws_size;
    const float* in   = (const float*)d_in[0];  // (16,32,256,256) f32
    const float* kern = (const float*)d_in[1];  // (32,5,5) f32
    float* out  = (float*)d_out;                // (16,32,256,256) f32
    float* xpad = (float*)d_ws;                 // 16*260*260 floats = 4.33 MB

    k_fill<<<((B * PIMG) / 4 + 255) / 256, 256, 0, stream>>>(xpad);
    k_cmax<<<(B * HWSZ / 4) / 256, 256, 0, stream>>>(in, xpad);
    dim3 grid(W / TILE, H / TILE, B);
    k_conv<<<grid, 256, 0, stream>>>(xpad, kern, out);
}